// PerExampleMaskParamsNet_30039001268704
// MI455X (gfx1250) — compile-verified
//
#include <hip/hip_runtime.h>

// CDNA5 / gfx1250 — wave32, WMMA (not MFMA).
typedef float v2f __attribute__((ext_vector_type(2)));
typedef float v4f __attribute__((ext_vector_type(4)));
typedef float v8f __attribute__((ext_vector_type(8)));

#define BATCH 32
#define D0_   1024
#define D1_   2048
#define D2_   1024

// sigmoid via CDNA5 V_TANH_F32: sigmoid(x) = 0.5*tanh(x/2) + 0.5
// (1 trans op instead of exp+rcp = 2)
__device__ __forceinline__ float fast_sigmoid(float x) {
    return __builtin_fmaf(0.5f, __builtin_amdgcn_tanhf(x * 0.5f), 0.5f);
}

// out[b, o] = act( bias[o] + sum_i sigmoid(alphas[b,o,i]) * W[o,i] * xin[b,i] )
// One block = 8 wave32s handles one (b, 16-output tile); waves split K, combine in LDS.
// WMMA mapping: A (16x4 f32) = masked-weight tile, B (4x16 f32) = x chunk broadcast
// across N; every column of D holds the same partial h, read from lanes 0 / 16.
// K=8 per b128 load via K-permutation: lane-half0 owns k0..k0+3, half1 owns k0+4..k0+7;
// WMMA#1 takes (e0,e1) of each half (logical K {k0,k0+1,k0+4,k0+5}), WMMA#2 takes
// (e2,e3). B uses the identical permutation, so the dot product is exact.
template <int DIN, int DOUT, int RELU>
__global__ __launch_bounds__(256) void masked_gemv_wmma(
    const float* __restrict__ alphas,  // [B, DOUT, DIN]
    const float* __restrict__ W,       // [DOUT, DIN]
    const float* __restrict__ bias,    // [DOUT]
    const float* __restrict__ xin,     // [B, DIN]
    float* __restrict__ out)           // [B, DOUT]
{
    __shared__ float xs[DIN];          // this example's input row
    __shared__ float hacc[16];         // cross-wave combine

    const int tid  = threadIdx.x;
    const int lane = tid & 31;
    const int wave = tid >> 5;
    const int b    = blockIdx.y;
    const int o0   = blockIdx.x << 4;

    // Stage this example's input row into LDS (vectorized b128 loads).
    {
        const float4* src = (const float4*)(xin + (size_t)b * DIN);
        float4* dst = (float4*)xs;
        #pragma unroll
        for (int i = tid; i < (DIN >> 2); i += 256) dst[i] = src[i];
    }
    if (tid < 16) hacc[tid] = 0.0f;
    __syncthreads();

    const int m   = lane & 15;           // A-tile row this lane feeds
    const int kk4 = (lane >> 4) << 2;    // 0 for lanes 0-15, 4 for lanes 16-31

    constexpr int KPW = DIN >> 3;        // K elements per wave (DIN / 8 waves)
    const int kbeg = wave * KPW;

    const float* arow = alphas + ((size_t)b * DOUT + (o0 + m)) * DIN + kk4;
    const float* wrow = W + (size_t)(o0 + m) * DIN + kk4;
    const float* xrow = xs + kk4;

    v8f acc0 = {};
    v8f acc1 = {};
    v8f acc2 = {};
    v8f acc3 = {};
    #pragma unroll 2
    for (int ku = 0; ku < KPW; ku += 16) {
        #pragma unroll
        for (int half = 0; half < 2; ++half) {
            const int k0 = kbeg + ku + half * 8;
            // alphas are single-use: stream non-temporally; W stays L2-resident.
            v4f aval = __builtin_nontemporal_load((const v4f*)(arow + k0));
            v4f wval = *(const v4f*)(wrow + k0);
            v4f xval = *(const v4f*)(xrow + k0);
            v4f mw;
            mw.x = wval.x * fast_sigmoid(aval.x);
            mw.y = wval.y * fast_sigmoid(aval.y);
            mw.z = wval.z * fast_sigmoid(aval.z);
            mw.w = wval.w * fast_sigmoid(aval.w);

            v2f a01; a01.x = mw.x;   a01.y = mw.y;
            v2f b01; b01.x = xval.x; b01.y = xval.y;
            v2f a23; a23.x = mw.z;   a23.y = mw.w;
            v2f b23; b23.x = xval.z; b23.y = xval.w;

            if (half == 0) {
                acc0 = __builtin_amdgcn_wmma_f32_16x16x4_f32(
                    false, a01, false, b01, (short)0, acc0, false, false);
                acc1 = __builtin_amdgcn_wmma_f32_16x16x4_f32(
                    false, a23, false, b23, (short)0, acc1, false, false);
            } else {
                acc2 = __builtin_amdgcn_wmma_f32_16x16x4_f32(
                    false, a01, false, b01, (short)0, acc2, false, false);
                acc3 = __builtin_amdgcn_wmma_f32_16x16x4_f32(
                    false, a23, false, b23, (short)0, acc3, false, false);
            }
        }
    }

    // D layout: VGPR r holds M=r (lanes 0-15) and M=8+r (lanes 16-31); all N equal.
    if (lane == 0 || lane == 16) {
        const int mbase = (lane >> 4) << 3;
        #pragma unroll
        for (int r = 0; r < 8; ++r)
            atomicAdd(&hacc[mbase + r], (acc0[r] + acc1[r]) + (acc2[r] + acc3[r]));
    }
    __syncthreads();

    if (tid < 16) {
        float v = hacc[tid] + bias[o0 + tid];
        if (RELU) v = fmaxf(v, 0.0f);
        out[(size_t)b * DOUT + o0 + tid] = v;
    }
}

extern "C" void kernel_launch(void* const* d_in, const int* in_sizes, int n_in,
                              void* d_out, int out_size, void* d_ws, size_t ws_size,
                              hipStream_t stream) {
    (void)in_sizes; (void)n_in; (void)out_size; (void)ws_size;
    const float* x  = (const float*)d_in[0];
    const float* W1 = (const float*)d_in[1];
    const float* b1 = (const float*)d_in[2];
    const float* W2 = (const float*)d_in[3];
    const float* b2 = (const float*)d_in[4];
    const float* a1 = (const float*)d_in[5];
    const float* a2 = (const float*)d_in[6];
    float* out = (float*)d_out;
    float* h   = (float*)d_ws;   // [BATCH, D1] f32 intermediate (256 KB)

    dim3 blk(256);
    // Layer 1: h = relu(sigmoid(a1)*W1 @ x + b1)
    masked_gemv_wmma<D0_, D1_, 1><<<dim3(D1_ / 16, BATCH), blk, 0, stream>>>(
        a1, W1, b1, x, h);
    // Layer 2: out = sigmoid(a2)*W2 @ h + b2
    masked_gemv_wmma<D1_, D2_, 0><<<dim3(D2_ / 16, BATCH), blk, 0, stream>>>(
        a2, W2, b2, h, out);
}